// GNN_84335977824921
// MI455X (gfx1250) — compile-verified
//
#include <hip/hip_runtime.h>

#define N_NODES  200000
#define N_EDGES  600000
#define N_GRAPHS 4000
#define EMB      128
#define LAYERS   5
#define TASKS    128
#define BN_EPS   1e-5f

typedef __attribute__((ext_vector_type(2))) float v2f;
typedef __attribute__((ext_vector_type(8))) float v8f;

// ---------------------------------------------------------------------------
// Generic fill
// ---------------------------------------------------------------------------
__global__ __launch_bounds__(256) void fill_kernel(float* __restrict__ p, float v, int n) {
    int i = blockIdx.x * blockDim.x + threadIdx.x;
    int stride = gridDim.x * blockDim.x;
    for (; i < n; i += stride) p[i] = v;
}

// ---------------------------------------------------------------------------
// Transpose a batch of 128x128 matrices: Wt[mat][n][k] = W[mat][k][n]
// ---------------------------------------------------------------------------
__global__ __launch_bounds__(256) void transpose128_kernel(
    const float* __restrict__ W, float* __restrict__ Wt, int nmat) {
    int t = blockIdx.x * blockDim.x + threadIdx.x;
    if (t >= nmat * EMB * EMB) return;
    int mat = t >> 14;
    int r   = (t >> 7) & 127;
    int c   = t & 127;
    Wt[(size_t)mat * EMB * EMB + (size_t)c * EMB + r] =
        W[(size_t)mat * EMB * EMB + (size_t)r * EMB + c];
}

// ---------------------------------------------------------------------------
// Atom encoder: h[n][:] = sum_f atom_emb[f][x[n][f]][:]
// one wave per node, 4 floats per lane
// ---------------------------------------------------------------------------
__global__ __launch_bounds__(256) void atom_encode_kernel(
    const int* __restrict__ x, const float* __restrict__ atom_emb,
    float* __restrict__ h) {
    int t = blockIdx.x * blockDim.x + threadIdx.x;
    int n = t >> 5;
    if (n >= N_NODES) return;
    int base = (t & 31) * 4;
    float4 acc = make_float4(0.f, 0.f, 0.f, 0.f);
#pragma unroll
    for (int f = 0; f < 9; ++f) {
        int v = x[n * 9 + f];
        const float4 e = *(const float4*)(atom_emb + ((size_t)(f * 64 + v)) * EMB + base);
        acc.x += e.x; acc.y += e.y; acc.z += e.z; acc.w += e.w;
    }
    *(float4*)(h + (size_t)n * EMB + base) = acc;
}

// ---------------------------------------------------------------------------
// Degree / norm precompute
// ---------------------------------------------------------------------------
__global__ __launch_bounds__(256) void degree_kernel(const int* __restrict__ src,
                                                     float* __restrict__ deg) {
    int e = blockIdx.x * blockDim.x + threadIdx.x;
    if (e >= N_EDGES) return;
    unsafeAtomicAdd(deg + src[e], 1.0f);
}

__global__ __launch_bounds__(256) void dinv_kernel(const float* __restrict__ deg,
                                                   float* __restrict__ dinv) {
    int n = blockIdx.x * blockDim.x + threadIdx.x;
    if (n < N_NODES) dinv[n] = rsqrtf(deg[n]);
}

__global__ __launch_bounds__(256) void norm_kernel(const int* __restrict__ src,
                                                   const int* __restrict__ dst,
                                                   const float* __restrict__ dinv,
                                                   float* __restrict__ nrm) {
    int e = blockIdx.x * blockDim.x + threadIdx.x;
    if (e < N_EDGES) nrm[e] = dinv[src[e]] * dinv[dst[e]];
}

// ---------------------------------------------------------------------------
// GEMM + bias via V_WMMA_F32_16X16X4_F32 (exact fp32).
// C[M,128] = A[M,128] * B[128,128] + bias[128], B passed TRANSPOSED (Bt[n][k]).
//
// Block = 256 threads = 8 waves; block handles one 16-row M-tile, wave w
// computes the 16x16 tile at column tile w. The shared 16x128 A-tile is
// staged once into LDS (coalesced float4) and read back as ds_load_b64
// fragments; B fragments are contiguous b64 loads from Bt.
//
// Layouts per CDNA5 ISA 7.12.2:
//   A 16x4 f32 (2 VGPR): lane = m%16 + 16*(k>=2), vgpr j -> k = 2*(lane/16)+j
//   B 4x16  f32 (2 VGPR): lane = n%16 + 16*half,   vgpr j -> k = 2*(lane/16)+j
//   C 16x16 f32 (8 VGPR): lane = n%16 + 16*half,   vgpr r -> m = r + 8*(lane/16)
// ---------------------------------------------------------------------------
__global__ __launch_bounds__(256) void gemm_bias_wmma(
    const float* __restrict__ A, const float* __restrict__ Bt,
    const float* __restrict__ bias, float* __restrict__ C, int M) {
    __shared__ float As[16 * EMB];  // 8 KB
    const int tid   = threadIdx.x;
    const int mTile = blockIdx.x;
    (void)M;

    // Stage A tile: 2048 floats, 2 x float4 per thread, fully coalesced.
    {
        const float4* s4 = (const float4*)(A + (size_t)mTile * 16 * EMB);
        float4* d4 = (float4*)As;
        d4[tid]       = s4[tid];
        d4[tid + 256] = s4[tid + 256];
    }
    __syncthreads();

    const int lane  = tid & 31;
    const int nTile = tid >> 5;   // 0..7
    const int hs    = lane >> 4;  // half-wave select
    const int l16   = lane & 15;
    const int n     = nTile * 16 + l16;   // B/C column held by this lane

    const float* Ap = As + l16 * EMB + 2 * hs;           // LDS, 8B aligned
    const float* Bp = Bt + (size_t)n * EMB + 2 * hs;     // global, 8B aligned
    v8f acc = {};
#pragma unroll
    for (int kk = 0; kk < EMB; kk += 4) {
        v2f a = *(const v2f*)(Ap + kk);   // ds_load_b64:     k = kk + 2*hs + {0,1}
        v2f b = *(const v2f*)(Bp + kk);   // global_load_b64: k = kk + 2*hs + {0,1}
        acc = __builtin_amdgcn_wmma_f32_16x16x4_f32(
            /*neg_a=*/false, a, /*neg_b=*/false, b,
            /*c_mod=*/(short)0, acc, /*reuse_a=*/false, /*reuse_b=*/false);
    }
    const float bv = bias[n];
    float* Cp = C + (size_t)(mTile * 16 + 8 * hs) * EMB + n;
#pragma unroll
    for (int r = 0; r < 8; ++r) Cp[(size_t)r * EMB] = acc[r] + bv;
}

// ---------------------------------------------------------------------------
// Edge message + scatter: agg[dst] += norm * relu(xl[src] + edge_emb)
// one wave per edge; 4 floats per lane; f32 global atomics
// ---------------------------------------------------------------------------
__global__ __launch_bounds__(256) void edge_msg_kernel(
    const int* __restrict__ src, const int* __restrict__ dst,
    const int* __restrict__ edge_attr, const float* __restrict__ nrm,
    const float* __restrict__ bond_emb_l,  // [3][8][128]
    const float* __restrict__ xl, float* __restrict__ agg) {
    int t = blockIdx.x * blockDim.x + threadIdx.x;
    int e = t >> 5;
    if (e >= N_EDGES) return;
    int base = (t & 31) * 4;

    int s = src[e], d = dst[e];
    float w = nrm[e];
    int a0 = edge_attr[e * 3 + 0];
    int a1 = edge_attr[e * 3 + 1];
    int a2 = edge_attr[e * 3 + 2];

    const float4 xv = *(const float4*)(xl + (size_t)s * EMB + base);
    const float4 e0 = *(const float4*)(bond_emb_l + (size_t)(0 * 8 + a0) * EMB + base);
    const float4 e1 = *(const float4*)(bond_emb_l + (size_t)(1 * 8 + a1) * EMB + base);
    const float4 e2 = *(const float4*)(bond_emb_l + (size_t)(2 * 8 + a2) * EMB + base);

    float4 msg;
    msg.x = w * fmaxf(xv.x + e0.x + e1.x + e2.x, 0.f);
    msg.y = w * fmaxf(xv.y + e0.y + e1.y + e2.y, 0.f);
    msg.z = w * fmaxf(xv.z + e0.z + e1.z + e2.z, 0.f);
    msg.w = w * fmaxf(xv.w + e0.w + e1.w + e2.w, 0.f);

    float* ap = agg + (size_t)d * EMB + base;
    unsafeAtomicAdd(ap + 0, msg.x);
    unsafeAtomicAdd(ap + 1, msg.y);
    unsafeAtomicAdd(ap + 2, msg.z);
    unsafeAtomicAdd(ap + 3, msg.w);
}

// ---------------------------------------------------------------------------
// Node update: h = BN(agg + relu(xl + root)/deg), optional relu
// ---------------------------------------------------------------------------
__global__ __launch_bounds__(256) void node_update_kernel(
    const float* __restrict__ agg, const float* __restrict__ xl,
    const float* __restrict__ root_l, const float* __restrict__ deg,
    const float* __restrict__ mean, const float* __restrict__ var,
    const float* __restrict__ gamma, const float* __restrict__ beta,
    float* __restrict__ h, int do_relu) {
    int t = blockIdx.x * blockDim.x + threadIdx.x;
    int n = t >> 5;
    if (n >= N_NODES) return;
    int base = (t & 31) * 4;
    float dinv = 1.0f / deg[n];

    const float4 a = *(const float4*)(agg + (size_t)n * EMB + base);
    const float4 x = *(const float4*)(xl + (size_t)n * EMB + base);
    float av[4] = {a.x, a.y, a.z, a.w};
    float xv[4] = {x.x, x.y, x.z, x.w};
    float out[4];
#pragma unroll
    for (int i = 0; i < 4; ++i) {
        int c = base + i;
        float r = fmaxf(xv[i] + root_l[c], 0.f);
        float v = av[i] + r * dinv;
        v = (v - mean[c]) * (gamma[c] * rsqrtf(var[c] + BN_EPS)) + beta[c];
        if (do_relu) v = fmaxf(v, 0.f);
        out[i] = v;
    }
    *(float4*)(h + (size_t)n * EMB + base) = make_float4(out[0], out[1], out[2], out[3]);
}

// ---------------------------------------------------------------------------
// Global add pool: hg[batch[n]] += h[n]
// ---------------------------------------------------------------------------
__global__ __launch_bounds__(256) void pool_kernel(
    const float* __restrict__ h, const int* __restrict__ batch,
    float* __restrict__ hg) {
    int t = blockIdx.x * blockDim.x + threadIdx.x;
    int n = t >> 5;
    if (n >= N_NODES) return;
    int base = (t & 31) * 4;
    int g = batch[n];
    const float4 v = *(const float4*)(h + (size_t)n * EMB + base);
    float* p = hg + (size_t)g * EMB + base;
    unsafeAtomicAdd(p + 0, v.x);
    unsafeAtomicAdd(p + 1, v.y);
    unsafeAtomicAdd(p + 2, v.z);
    unsafeAtomicAdd(p + 3, v.w);
}

// ---------------------------------------------------------------------------
extern "C" void kernel_launch(void* const* d_in, const int* in_sizes, int n_in,
                              void* d_out, int out_size, void* d_ws, size_t ws_size,
                              hipStream_t stream) {
    const int*   x         = (const int*)d_in[0];
    const int*   edge_idx  = (const int*)d_in[1];
    const int*   src       = edge_idx;
    const int*   dst       = edge_idx + N_EDGES;
    const int*   edge_attr = (const int*)d_in[2];
    const int*   batch     = (const int*)d_in[3];
    const float* atom_emb  = (const float*)d_in[4];
    const float* bond_emb  = (const float*)d_in[5];
    const float* W         = (const float*)d_in[6];
    const float* b         = (const float*)d_in[7];
    const float* root      = (const float*)d_in[8];
    const float* bn_mean   = (const float*)d_in[9];
    const float* bn_var    = (const float*)d_in[10];
    const float* bn_gamma  = (const float*)d_in[11];
    const float* bn_beta   = (const float*)d_in[12];
    const float* headW     = (const float*)d_in[13];
    const float* headb     = (const float*)d_in[14];
    float*       out       = (float*)d_out;
    (void)in_sizes; (void)n_in; (void)out_size; (void)ws_size;

    char* ws = (char*)d_ws;
    size_t off = 0;
    auto carve = [&](size_t bytes) -> float* {
        float* p = (float*)(ws + off);
        off = (off + bytes + 255) & ~(size_t)255;
        return p;
    };
    const size_t nodeMatBytes = (size_t)N_NODES * EMB * sizeof(float);
    float* h      = carve(nodeMatBytes);
    float* xl     = carve(nodeMatBytes);
    float* agg    = carve(nodeMatBytes);
    float* deg    = carve((size_t)N_NODES * sizeof(float));
    float* dinv   = carve((size_t)N_NODES * sizeof(float));
    float* nrm    = carve((size_t)N_EDGES * sizeof(float));
    float* hg     = carve((size_t)N_GRAPHS * EMB * sizeof(float));
    float* Wt     = carve((size_t)LAYERS * EMB * EMB * sizeof(float));
    float* headWt = carve((size_t)EMB * EMB * sizeof(float));

    const int nodeWaveBlocks = (N_NODES * 32 + 255) / 256;   // 25000
    const int edgeWaveBlocks = (N_EDGES * 32 + 255) / 256;   // 75000
    const int edgeBlocks     = (N_EDGES + 255) / 256;
    const int nodeBlocks     = (N_NODES + 255) / 256;

    // One-time transposes so GEMM B fragments are contiguous b64 loads.
    transpose128_kernel<<<(LAYERS * EMB * EMB + 255) / 256, 256, 0, stream>>>(W, Wt, LAYERS);
    transpose128_kernel<<<(EMB * EMB + 255) / 256, 256, 0, stream>>>(headW, headWt, 1);

    // h = atom encoder
    atom_encode_kernel<<<nodeWaveBlocks, 256, 0, stream>>>(x, atom_emb, h);

    // degree / dinv / norm
    fill_kernel<<<2048, 256, 0, stream>>>(deg, 1.0f, N_NODES);
    degree_kernel<<<edgeBlocks, 256, 0, stream>>>(src, deg);
    dinv_kernel<<<nodeBlocks, 256, 0, stream>>>(deg, dinv);
    norm_kernel<<<edgeBlocks, 256, 0, stream>>>(src, dst, dinv, nrm);

    for (int l = 0; l < LAYERS; ++l) {
        const float* Wtl   = Wt + (size_t)l * EMB * EMB;
        const float* bl    = b + (size_t)l * EMB;
        const float* rootl = root + (size_t)l * EMB;
        const float* bondl = bond_emb + (size_t)l * 3 * 8 * EMB;

        // xl = h @ W[l] + b[l]   (WMMA f32, LDS-staged A, transposed B)
        gemm_bias_wmma<<<N_NODES / 16, 256, 0, stream>>>(h, Wtl, bl, xl, N_NODES);

        // agg = segment_sum over edges
        fill_kernel<<<2048, 256, 0, stream>>>(agg, 0.0f, N_NODES * EMB);
        edge_msg_kernel<<<edgeWaveBlocks, 256, 0, stream>>>(
            src, dst, edge_attr, nrm, bondl, xl, agg);

        // h = BN(agg + relu(xl + root)/deg) [+ relu]
        node_update_kernel<<<nodeWaveBlocks, 256, 0, stream>>>(
            agg, xl, rootl, deg,
            bn_mean + (size_t)l * EMB, bn_var + (size_t)l * EMB,
            bn_gamma + (size_t)l * EMB, bn_beta + (size_t)l * EMB,
            h, (l != LAYERS - 1) ? 1 : 0);
    }

    // global add pool
    fill_kernel<<<2048, 256, 0, stream>>>(hg, 0.0f, N_GRAPHS * EMB);
    pool_kernel<<<nodeWaveBlocks, 256, 0, stream>>>(h, batch, hg);

    // head: out = hg @ headW + headb   (WMMA f32)
    gemm_bias_wmma<<<N_GRAPHS / 16, 256, 0, stream>>>(hg, headWt, headb, out, N_GRAPHS);
}